// TensoRF_Slim_69423851373269
// MI455X (gfx1250) — compile-verified
//
#include <hip/hip_runtime.h>
#include <hip/hip_bf16.h>
#include <math.h>

typedef __attribute__((ext_vector_type(2))) float v2f;
typedef __attribute__((ext_vector_type(8))) float v8f;

namespace {

constexpr int NR = 4096;
constexpr int NS = 256;
constexpr int NSAMP = NR * NS;   // 1,048,576
constexpr int G = 300;
constexpr int DC = 8;            // density channels per plane
constexpr int AC = 24;           // appearance channels per plane
constexpr float DENSITY_SHIFT = -10.0f;
constexpr float DISTANCE_SCALE = 25.0f;

struct Bilin { int base, dx, dy; float w00, w01, w10, w11; };
struct Lin   { int i0, d; float w; };

__device__ __forceinline__ Bilin setup2d(float cx, float cy) {
    float x = (cx + 1.0f) * 0.5f * (float)(G - 1);
    float y = (cy + 1.0f) * 0.5f * (float)(G - 1);
    float xf = floorf(x), yf = floorf(y);
    float wx = x - xf,    wy = y - yf;
    int x0 = (int)xf; x0 = x0 < 0 ? 0 : (x0 > G - 1 ? G - 1 : x0);
    int y0 = (int)yf; y0 = y0 < 0 ? 0 : (y0 > G - 1 ? G - 1 : y0);
    int x1 = (x0 + 1 > G - 1) ? G - 1 : x0 + 1;
    int y1 = (y0 + 1 > G - 1) ? G - 1 : y0 + 1;
    Bilin b;
    b.base = y0 * G + x0;
    b.dx   = x1 - x0;
    b.dy   = (y1 - y0) * G;
    b.w00 = (1.0f - wx) * (1.0f - wy);
    b.w01 = wx * (1.0f - wy);
    b.w10 = (1.0f - wx) * wy;
    b.w11 = wx * wy;
    return b;
}

__device__ __forceinline__ float sample2d(const float* __restrict__ p, const Bilin& b) {
    float v00 = p[b.base];
    float v01 = p[b.base + b.dx];
    float v10 = p[b.base + b.dy];
    float v11 = p[b.base + b.dy + b.dx];
    return v00 * b.w00 + v01 * b.w01 + v10 * b.w10 + v11 * b.w11;
}

__device__ __forceinline__ Lin setup1d(float t) {
    float y = (t + 1.0f) * 0.5f * (float)(G - 1);
    float yf = floorf(y);
    int i0 = (int)yf; i0 = i0 < 0 ? 0 : (i0 > G - 1 ? G - 1 : i0);
    int i1 = (i0 + 1 > G - 1) ? G - 1 : i0 + 1;
    Lin l; l.i0 = i0; l.d = i1 - i0; l.w = y - yf;
    return l;
}

__device__ __forceinline__ float sample1d(const float* __restrict__ p, const Lin& l) {
    float a = p[l.i0];
    float b = p[l.i0 + l.d];
    return a + l.w * (b - a);
}

__device__ __forceinline__ float softplus_f(float x) {
    // jax.nn.softplus: log1p(exp(x)) with overflow guard
    return (x > 30.0f) ? x : log1pf(expf(x));
}

// ---------------------------------------------------------------------------
// Kernel 1: per-sample TensoRF features.
//   One wave owns 16 samples. lane L: sample tile_base + (L&15), K-half h=L>>4.
//   sigma_feat -> sigma_ws[s]; rgb via V_WMMA_F32_16X16X4_F32 -> rgb_ws[r*NSAMP+s].
// ---------------------------------------------------------------------------
__global__ __launch_bounds__(256) void tensorf_feat_kernel(
    const float* __restrict__ xyz_sampled,   // [NR*NS, 3]
    const float* __restrict__ dpl,           // [3, DC, G, G]
    const float* __restrict__ dln,           // [3, DC, G]
    const float* __restrict__ apl,           // [3, AC, G, G]
    const float* __restrict__ aln,           // [3, AC, G]
    const float* __restrict__ dbw,           // [1, 24]
    const float* __restrict__ abw,           // [3, 72]
    const float* __restrict__ aabb,          // [2, 3]
    float* __restrict__ sigma_ws,            // [NSAMP]
    float* __restrict__ rgb_ws)              // [3, NSAMP]
{
    const int lane = threadIdx.x & 31;
    const int wave = blockIdx.x * (blockDim.x >> 5) + (threadIdx.x >> 5);
    const int h = lane >> 4;     // K-half (0/1)
    const int m = lane & 15;     // A row / sample column within tile
    const int s = wave * 16 + m; // global sample id

    // normalized coordinates
    float lo0 = aabb[0], lo1 = aabb[1], lo2 = aabb[2];
    float inv0 = 2.0f / (aabb[3] - lo0);
    float inv1 = 2.0f / (aabb[4] - lo1);
    float inv2 = 2.0f / (aabb[5] - lo2);
    float p0 = (xyz_sampled[s * 3 + 0] - lo0) * inv0 - 1.0f;
    float p1 = (xyz_sampled[s * 3 + 1] - lo1) * inv1 - 1.0f;
    float p2 = (xyz_sampled[s * 3 + 2] - lo2) * inv2 - 1.0f;

    // MAT_MODE = ((0,1),(0,2),(1,2)); VEC_MODE = (2,1,0)
    const float cxs[3] = { p0, p0, p1 };
    const float cys[3] = { p1, p2, p2 };
    const float tvs[3] = { p2, p1, p0 };

    Bilin bb[3]; Lin ll[3];
#pragma unroll
    for (int i = 0; i < 3; ++i) { bb[i] = setup2d(cxs[i], cys[i]); ll[i] = setup1d(tvs[i]); }

    // ---------------- density: this half-lane handles channels c%4 in {2h,2h+1}
    float dsum = 0.0f;
#pragma unroll
    for (int i = 0; i < 3; ++i) {
        const float* pl = dpl + i * (DC * G * G);
        const float* ln = dln + i * (DC * G);
#pragma unroll
        for (int q = 0; q < 2; ++q) {
#pragma unroll
            for (int e = 0; e < 2; ++e) {
                int c = 4 * q + 2 * h + e;
                float f = sample2d(pl + c * (G * G), bb[i]) * sample1d(ln + c * G, ll[i]);
                dsum += dbw[i * DC + c] * f;
            }
        }
    }
    dsum += __shfl_xor(dsum, 16, 32);

    // ---------------- appearance via WMMA f32 16x16x4, K = 72 = 18 fragments
    v2f a[18];
#pragma unroll
    for (int j = 0; j < 18; ++j) {
        v2f af; af.x = 0.0f; af.y = 0.0f;
        if (m < 3) {
            af.x = abw[m * 72 + 4 * j + 2 * h + 0];
            af.y = abw[m * 72 + 4 * j + 2 * h + 1];
        }
        a[j] = af;
    }

    v8f acc = {};
#pragma unroll
    for (int i = 0; i < 3; ++i) {
        const float* pl = apl + i * (AC * G * G);
        const float* ln = aln + i * (AC * G);
#pragma unroll
        for (int q = 0; q < 6; ++q) {
            int c0 = 4 * q + 2 * h;
            v2f bf;
            bf.x = sample2d(pl + (c0 + 0) * (G * G), bb[i]) * sample1d(ln + (c0 + 0) * G, ll[i]);
            bf.y = sample2d(pl + (c0 + 1) * (G * G), bb[i]) * sample1d(ln + (c0 + 1) * G, ll[i]);
            // D = A(3x72 padded to 16) x B(72x16 samples) + C
            acc = __builtin_amdgcn_wmma_f32_16x16x4_f32(
                false, a[i * 6 + q], false, bf, (short)0, acc, false, false);
        }
    }

    // D layout: lanes 0-15 hold M = VGPR index; rows 0..2 are R,G,B
    if (h == 0) {
        sigma_ws[s] = dsum;
        rgb_ws[0 * NSAMP + s] = acc[0];
        rgb_ws[1 * NSAMP + s] = acc[1];
        rgb_ws[2 * NSAMP + s] = acc[2];
    }
}

// ---------------------------------------------------------------------------
// Kernel 2: wave-per-ray volumetric compositing (segmented prefix product).
// ---------------------------------------------------------------------------
__global__ __launch_bounds__(256) void tensorf_composite_kernel(
    const float* __restrict__ zvals,     // [NR, NS]
    const float* __restrict__ sigma_ws,  // [NSAMP]
    const float* __restrict__ rgb_ws,    // [3, NSAMP]
    float* __restrict__ out)             // [NR, 3]
{
    const int lane = threadIdx.x & 31;
    const int ray  = blockIdx.x * (blockDim.x >> 5) + (threadIdx.x >> 5);
    if (ray >= NR) return;

    const float* z = zvals + ray * NS;
    float carry = 1.0f;
    float accr = 0.0f, accg = 0.0f, accb = 0.0f;

#pragma unroll
    for (int t = 0; t < NS / 32; ++t) {
        int si = t * 32 + lane;
        float dist = (si < NS - 1) ? (z[si + 1] - z[si]) : (z[NS - 1] - z[NS - 2]);
        int sidx = ray * NS + si;

        float sig = softplus_f(sigma_ws[sidx] + DENSITY_SHIFT);
        float e = expf(-sig * dist * DISTANCE_SCALE);
        float alpha = 1.0f - e;
        float u = (1.0f - alpha) + 1e-10f;

        // inclusive prefix product of u across the wave
        float p = u;
#pragma unroll
        for (int off = 1; off < 32; off <<= 1) {
            float v = __shfl_up(p, off, 32);
            if (lane >= off) p *= v;
        }
        float pexcl = __shfl_up(p, 1, 32);
        if (lane == 0) pexcl = 1.0f;

        float w = alpha * carry * pexcl;
        accr += w * rgb_ws[0 * NSAMP + sidx];
        accg += w * rgb_ws[1 * NSAMP + sidx];
        accb += w * rgb_ws[2 * NSAMP + sidx];

        carry *= __shfl(p, 31, 32);
    }

    // cross-lane reduction
#pragma unroll
    for (int off = 16; off >= 1; off >>= 1) {
        accr += __shfl_xor(accr, off, 32);
        accg += __shfl_xor(accg, off, 32);
        accb += __shfl_xor(accb, off, 32);
    }
    if (lane == 0) {
        out[ray * 3 + 0] = accr;
        out[ray * 3 + 1] = accg;
        out[ray * 3 + 2] = accb;
    }
}

} // namespace

extern "C" void kernel_launch(void* const* d_in, const int* in_sizes, int n_in,
                              void* d_out, int out_size, void* d_ws, size_t ws_size,
                              hipStream_t stream) {
    const float* xyz   = (const float*)d_in[0];
    // d_in[1] = viewdirs (unused by the reference math path)
    const float* zvals = (const float*)d_in[2];
    const float* dpl   = (const float*)d_in[3];
    const float* dln   = (const float*)d_in[4];
    const float* apl   = (const float*)d_in[5];
    const float* aln   = (const float*)d_in[6];
    const float* dbw   = (const float*)d_in[7];
    const float* abw   = (const float*)d_in[8];
    const float* aabb  = (const float*)d_in[9];

    float* sigma_ws = (float*)d_ws;          // NSAMP floats
    float* rgb_ws   = sigma_ws + NSAMP;      // 3*NSAMP floats

    // 16 samples per wave, 8 waves per block -> 128 samples/block
    const int feat_blocks = NSAMP / 128;     // 8192
    tensorf_feat_kernel<<<feat_blocks, 256, 0, stream>>>(
        xyz, dpl, dln, apl, aln, dbw, abw, aabb, sigma_ws, rgb_ws);

    const int comp_blocks = NR / 8;          // 512 (8 rays/block, wave per ray)
    tensorf_composite_kernel<<<comp_blocks, 256, 0, stream>>>(
        zvals, sigma_ws, rgb_ws, (float*)d_out);
}